// MultiScaleAttention_19928648253537
// MI455X (gfx1250) — compile-verified
//
#include <hip/hip_runtime.h>
#include <math.h>

// ---------------------------------------------------------------------------
// MultiScaleAttention for MI455X (gfx1250), fp32 end-to-end.
// GEMMs on V_WMMA_F32_16X16X4_F32; global->LDS staging via the gfx1250
// async-copy path (GLOBAL_LOAD_ASYNC_TO_LDS_B128) when the builtin exists,
// else vectorized b128 load/store.
// ---------------------------------------------------------------------------

typedef __attribute__((ext_vector_type(2))) float v2f;
typedef __attribute__((ext_vector_type(4))) float v4f;
typedef __attribute__((ext_vector_type(8))) float v8f;
typedef int ai4 __attribute__((__vector_size__(16)));   // int4, matches builtin

#define BATCH 8
#define SEQ   1024
#define CDIM  768
#define C3    2304
#define HEADS 12
#define HD    64

#define AS1 __attribute__((address_space(1)))
#define AS3 __attribute__((address_space(3)))

#if __has_builtin(__builtin_amdgcn_global_load_async_to_lds_b128)
#define ASYNC_B128 __builtin_amdgcn_global_load_async_to_lds_b128
#define HAVE_ASYNC_LDS 1
#elif __has_builtin(__builtin_amdgcn_global_async_load_to_lds_b128)
#define ASYNC_B128 __builtin_amdgcn_global_async_load_to_lds_b128
#define HAVE_ASYNC_LDS 1
#endif

__device__ __forceinline__ void cp_f4(float* l, const float* g) {
#ifdef HAVE_ASYNC_LDS
  ASYNC_B128((AS1 ai4*)(g), (AS3 ai4*)(l), 0, 0);
#else
  *(v4f*)l = *(const v4f*)g;
#endif
}

__device__ __forceinline__ void async_wait() {
#ifdef HAVE_ASYNC_LDS
#if __has_builtin(__builtin_amdgcn_s_wait_asynccnt)
  __builtin_amdgcn_s_wait_asynccnt(0);
#else
  asm volatile("s_wait_asynccnt 0x0" ::: "memory");
#endif
  asm volatile("" ::: "memory");
#endif
}

__device__ __forceinline__ v8f wmma_f32(v2f a, v2f b, v8f c) {
  // (neg_a, A, neg_b, B, c_mod, C, reuse_a, reuse_b)
  return __builtin_amdgcn_wmma_f32_16x16x4_f32(false, a, false, b, (short)0, c,
                                               false, false);
}

__device__ __forceinline__ float fusion_weight(const float* fw, int idx) {
  float f0 = fw[0], f1 = fw[1], f2 = fw[2];
  float m = fmaxf(f0, fmaxf(f1, f2));
  float e0 = __expf(f0 - m), e1 = __expf(f1 - m), e2 = __expf(f2 - m);
  float s = e0 + e1 + e2;
  float e = (idx == 0) ? e0 : ((idx == 1) ? e1 : e2);
  return e / s;
}

// ---------------------------------------------------------------------------
// Tiled fp32 WMMA GEMM:  D[M,N] = alpha * (A[M,K] @ W[K,N] + bias[N])  (+= opt)
// Block = 128 threads (4 waves), tile = 64x64, each wave owns a 32x32 subtile
// as 2x2 WMMA 16x16 fragments. M,N multiples of 64; K multiple of 16.
// A LDS pitch = 20 floats (80B: b128-aligned, 20*l16 mod 64 hits 16 banks).
// ---------------------------------------------------------------------------
__launch_bounds__(128)
__global__ void wmma_gemm_kernel(const float* __restrict__ A,
                                 const float* __restrict__ W,
                                 const float* __restrict__ bias,
                                 float* __restrict__ D,
                                 int M, int N, int K,
                                 const float* fw, int fwidx, int accum) {
  __shared__ __align__(16) float As[64][20];
  __shared__ __align__(16) float Bs[16][64];

  const int tid  = threadIdx.x;
  const int lane = tid & 31;
  const int wv   = tid >> 5;
  const int wm   = wv >> 1, wn = wv & 1;
  const int hi   = (lane >> 4) & 1;   // lanes 16..31 hold K+2 / M+8 halves
  const int l16  = lane & 15;
  const int m0   = blockIdx.y * 64;
  const int n0   = blockIdx.x * 64;

  v8f acc[2][2] = {};

  for (int k0 = 0; k0 < K; k0 += 16) {
    __syncthreads();
#pragma unroll
    for (int t = 0; t < 2; t++) {            // A tile 64x16 (f4 chunks)
      int idx = tid + t * 128;
      int r = idx >> 2, c = (idx & 3) * 4;
      cp_f4(&As[r][c], &A[(size_t)(m0 + r) * K + k0 + c]);
    }
#pragma unroll
    for (int t = 0; t < 2; t++) {            // W tile 16x64 (f4 chunks)
      int idx = tid + t * 128;
      int r = idx >> 4, c = (idx & 15) * 4;
      cp_f4(&Bs[r][c], &W[(size_t)(k0 + r) * N + n0 + c]);
    }
    if (k0 + 16 < K) {                       // global_prefetch_b8 path
      __builtin_prefetch(&A[(size_t)(m0 + (tid >> 1)) * K + k0 + 16], 0, 3);
      __builtin_prefetch(&W[(size_t)(k0 + 16 + (tid >> 6)) * N + n0 + (tid & 63)], 0, 3);
    }
    async_wait();
    __syncthreads();
#pragma unroll
    for (int kk = 0; kk < 4; kk++) {
      int kb = kk * 4 + hi * 2;
      v2f a[2], bf[2];
#pragma unroll
      for (int i = 0; i < 2; i++) {
        int r = wm * 32 + i * 16 + l16;      // A: lane holds row M=l16, K pair
        a[i].x = As[r][kb];
        a[i].y = As[r][kb + 1];
      }
#pragma unroll
      for (int j = 0; j < 2; j++) {
        int c = wn * 32 + j * 16 + l16;      // B: lane holds col N=l16, K pair
        bf[j].x = Bs[kb][c];
        bf[j].y = Bs[kb + 1][c];
      }
#pragma unroll
      for (int i = 0; i < 2; i++)
#pragma unroll
        for (int j = 0; j < 2; j++)
          acc[i][j] = wmma_f32(a[i], bf[j], acc[i][j]);
    }
  }

  float alpha = fw ? fusion_weight(fw, fwidx) : 1.0f;
#pragma unroll
  for (int i = 0; i < 2; i++) {
#pragma unroll
    for (int j = 0; j < 2; j++) {
      int col = n0 + wn * 32 + j * 16 + l16;
      float bv = bias ? bias[col] : 0.0f;
#pragma unroll
      for (int r = 0; r < 8; r++) {          // C frag: M = r + 8*hi
        int row = m0 + wm * 32 + i * 16 + r + hi * 8;
        float v = (acc[i][j][r] + bv) * alpha;
        size_t off = (size_t)row * N + col;
        if (accum) D[off] += v; else D[off] = v;
      }
    }
  }
}

// ---------------------------------------------------------------------------
// Attention: one workgroup per (batch, 16-query-row tile); loops over 12 heads.
// qkv layout: [B*Ns, 2304], q at col h*64, k at 768+h*64, v at 1536+h*64.
// Scores + softmax + head-mean kept in LDS (deterministic, no atomics).
// Q/K staging pitch = 76 floats (304B: b128-aligned, 76*l16 mod 64 -> 16 banks)
// ---------------------------------------------------------------------------
#define QKP 76
__launch_bounds__(128)
__global__ void attn_kernel(const float* __restrict__ qkv,
                            float* __restrict__ obuf,
                            float* __restrict__ attn_out,
                            int Ns) {
  extern __shared__ float sm[];
  const int tid  = threadIdx.x;
  const int lane = tid & 31;
  const int wv   = tid >> 5;
  const int hi   = (lane >> 4) & 1;
  const int l16  = lane & 15;
  const int b    = blockIdx.y;
  const int q0   = blockIdx.x * 16;
  const int T    = Ns >> 4;       // key tiles
  const int sp   = Ns + 4;        // padded score pitch

  float* q_lds   = sm;                        // 16*QKP
  float* kst     = q_lds + 16 * QKP;          // 4 waves * 16*QKP
  float* vst     = kst + 4 * 16 * QKP;        // 16*64
  float* s_lds   = vst + 16 * 64;             // 16*sp
  float* m_lds   = s_lds + 16 * sp;           // 16*Ns
  float* red     = m_lds + 16 * Ns;           // 128
  float* rowstat = red + 128;                 // 32

  for (int i = tid; i < 16 * Ns; i += 128) m_lds[i] = 0.0f;

  float* kw = kst + wv * 16 * QKP;

  for (int h = 0; h < HEADS; h++) {
    __syncthreads();  // guards prev head's s_lds readers vs new writes
#pragma unroll
    for (int t = 0; t < 2; t++) {             // Q tile 16x64 -> LDS (f4)
      int idx = tid + t * 128;
      int r = idx >> 4, c = (idx & 15) * 4;
      cp_f4(&q_lds[r * QKP + c], &qkv[(size_t)(b * Ns + q0 + r) * C3 + h * HD + c]);
    }
    async_wait();
    __syncthreads();

    // ---- scores: S[16,Ns] = (Q Kt) * d^-0.5 ; waves split key tiles ----
    for (int kt = wv; kt < T; kt += 4) {
#pragma unroll
      for (int t = 0; t < 8; t++) {           // stage K tile (wave-private, f4)
        int idx = lane + t * 32;
        int r = idx >> 4, c = (idx & 15) * 4;
        cp_f4(&kw[r * QKP + c],
              &qkv[(size_t)(b * Ns + kt * 16 + r) * C3 + 768 + h * HD + c]);
      }
      async_wait();
      v8f acc = {};
#pragma unroll
      for (int kk = 0; kk < 16; kk++) {       // K=64 -> 16 wmma k4-steps
        int kb = kk * 4 + hi * 2;
        v2f a, bb;
        a.x  = q_lds[l16 * QKP + kb];  a.y  = q_lds[l16 * QKP + kb + 1];
        bb.x = kw[l16 * QKP + kb];     bb.y = kw[l16 * QKP + kb + 1];
        acc = wmma_f32(a, bb, acc);
      }
#pragma unroll
      for (int r = 0; r < 8; r++)
        s_lds[(r + hi * 8) * sp + kt * 16 + l16] = acc[r] * 0.125f;  // 64^-0.5
    }
    __syncthreads();

    // ---- softmax over each of 16 rows (8 threads per row) ----
    {
      int row = tid >> 3, sub = tid & 7;
      float* srow = s_lds + row * sp;
      float mx = -3.0e38f;
      for (int c = sub; c < Ns; c += 8) mx = fmaxf(mx, srow[c]);
      red[tid] = mx;
      __syncthreads();
      if (sub == 0) {
        float m2 = red[row * 8];
        for (int k2 = 1; k2 < 8; k2++) m2 = fmaxf(m2, red[row * 8 + k2]);
        rowstat[row] = m2;
      }
      __syncthreads();
      float rmax = rowstat[row];
      float ssum = 0.0f;
      for (int c = sub; c < Ns; c += 8) {
        float e = __expf(srow[c] - rmax);
        srow[c] = e;
        ssum += e;
      }
      red[tid] = ssum;
      __syncthreads();
      if (sub == 0) {
        float s2 = 0.0f;
        for (int k2 = 0; k2 < 8; k2++) s2 += red[row * 8 + k2];
        rowstat[16 + row] = 1.0f / s2;
      }
      __syncthreads();
      float inv = rowstat[16 + row];
      float* mrow = m_lds + row * Ns;
      for (int c = sub; c < Ns; c += 8) {
        float p = srow[c] * inv;
        srow[c] = p;
        mrow[c] += p * (1.0f / (float)HEADS);
      }
    }

    // ---- O[16,64] = P[16,Ns] @ V[Ns,64] ; wave wv owns output cols wv*16.. --
    v8f oacc = {};
    for (int kt = 0; kt < T; kt++) {
      __syncthreads();
#pragma unroll
      for (int t = 0; t < 2; t++) {           // stage V tile 16x64 (shared, f4)
        int idx = tid + t * 128;
        int r = idx >> 4, c = (idx & 15) * 4;
        cp_f4(&vst[r * 64 + c],
              &qkv[(size_t)(b * Ns + kt * 16 + r) * C3 + 1536 + h * HD + c]);
      }
      async_wait();
      __syncthreads();
#pragma unroll
      for (int kk = 0; kk < 4; kk++) {
        int kb = kt * 16 + kk * 4 + hi * 2;
        int krow = kk * 4 + hi * 2;
        v2f a, bb;
        a.x = s_lds[l16 * sp + kb];
        a.y = s_lds[l16 * sp + kb + 1];
        bb.x = vst[krow * 64 + wv * 16 + l16];
        bb.y = vst[(krow + 1) * 64 + wv * 16 + l16];
        oacc = wmma_f32(a, bb, oacc);
      }
    }
#pragma unroll
    for (int r = 0; r < 8; r++) {
      int row = q0 + r + hi * 8;
      obuf[(size_t)(b * Ns + row) * CDIM + h * HD + wv * 16 + l16] = oacc[r];
    }
  }

  __syncthreads();
  for (int i = tid; i < 16 * Ns; i += 128) {   // head-mean attention out
    int r = i / Ns, c = i % Ns;
    attn_out[((size_t)b * Ns + q0 + r) * Ns + c] = m_lds[r * Ns + c];
  }
}

// ---------------------------------------------------------------------------
// s x s strided average pool:  x[B,1024,768] -> xs[B,(32/s)^2,768]
// ---------------------------------------------------------------------------
__global__ void downsample_kernel(const float* __restrict__ x,
                                  float* __restrict__ xs, int s, int hs) {
  int total = BATCH * hs * hs * CDIM;
  int idx = blockIdx.x * 256 + threadIdx.x;
  if (idx >= total) return;
  int c = idx % CDIM;
  int n = (idx / CDIM) % (hs * hs);
  int b = idx / (CDIM * hs * hs);
  int i = n / hs, j = n % hs;
  float acc = 0.0f;
  for (int di = 0; di < s; di++)
    for (int dj = 0; dj < s; dj++)
      acc += x[((size_t)b * SEQ + (i * s + di) * 32 + (j * s + dj)) * CDIM + c];
  xs[idx] = acc / (float)(s * s);
}

// ---------------------------------------------------------------------------
// Half-pixel bilinear upsample (hs x hs -> 32 x 32), fused += w_i * up(feat)
// ---------------------------------------------------------------------------
__global__ void upsample_add_kernel(const float* __restrict__ small,
                                    float* __restrict__ fused,
                                    const float* __restrict__ fw, int fwidx,
                                    int hs) {
  int total = BATCH * SEQ * CDIM;
  int idx = blockIdx.x * 256 + threadIdx.x;
  if (idx >= total) return;
  int c = idx % CDIM;
  int n = (idx / CDIM) % SEQ;
  int b = idx / (CDIM * SEQ);
  int y = n / 32, x = n % 32;
  float scale = (float)hs / 32.0f;
  float fy = (y + 0.5f) * scale - 0.5f;
  float fx = (x + 0.5f) * scale - 0.5f;
  int y0 = (int)floorf(fy), x0 = (int)floorf(fx);
  float wy = fy - (float)y0, wx = fx - (float)x0;
  int y0c = min(max(y0, 0), hs - 1), y1c = min(max(y0 + 1, 0), hs - 1);
  int x0c = min(max(x0, 0), hs - 1), x1c = min(max(x0 + 1, 0), hs - 1);
  const float* sb = small + (size_t)b * hs * hs * CDIM;
  float v00 = sb[(size_t)(y0c * hs + x0c) * CDIM + c];
  float v01 = sb[(size_t)(y0c * hs + x1c) * CDIM + c];
  float v10 = sb[(size_t)(y1c * hs + x0c) * CDIM + c];
  float v11 = sb[(size_t)(y1c * hs + x1c) * CDIM + c];
  float v = (1.0f - wy) * ((1.0f - wx) * v00 + wx * v01) +
            wy * ((1.0f - wx) * v10 + wx * v11);
  fused[idx] += fusion_weight(fw, fwidx) * v;
}

// ---------------------------------------------------------------------------
extern "C" void kernel_launch(void* const* d_in, const int* in_sizes, int n_in,
                              void* d_out, int out_size, void* d_ws,
                              size_t ws_size, hipStream_t stream) {
  const float* x      = (const float*)d_in[0];
  const float* qkv_w  = (const float*)d_in[1];
  const float* qkv_b  = (const float*)d_in[2];
  const float* proj_w = (const float*)d_in[3];
  const float* proj_b = (const float*)d_in[4];
  const float* sp_w   = (const float*)d_in[5];
  const float* sp_b   = (const float*)d_in[6];
  const float* fw     = (const float*)d_in[7];
  const float* out_w  = (const float*)d_in[8];
  const float* out_b  = (const float*)d_in[9];

  // workspace layout (floats): ~163.6 MB total
  float* ws      = (float*)d_ws;
  float* qkv_buf = ws;                                   // 8192*2304
  float* o_buf   = qkv_buf + (size_t)8192 * 2304;        // 8192*768
  float* att_buf = o_buf   + (size_t)8192 * 768;         // 8192*768
  float* fused   = att_buf + (size_t)8192 * 768;         // 8192*768
  float* xs_buf  = fused   + (size_t)8192 * 768;         // 2048*768
  float* sm_buf  = xs_buf  + (size_t)2048 * 768;         // 2048*768

  // output layout: out | attn0 | attn1 | attn2
  float* out   = (float*)d_out;
  float* attn0 = out + (size_t)BATCH * SEQ * CDIM;
  float* attn1 = attn0 + (size_t)BATCH * 1024 * 1024;
  float* attn2 = attn1 + (size_t)BATCH * 256 * 256;

  dim3 blk(128);

  for (int i = 0; i < 3; i++) {
    int s  = (i == 0) ? 1 : (i == 1) ? 2 : 4;
    int hs = 32 / s;
    int Ns = hs * hs;
    int Msub = BATCH * Ns;

    const float* xin = x;
    if (s != 1) {
      int tot = BATCH * Ns * CDIM;
      downsample_kernel<<<(tot + 255) / 256, 256, 0, stream>>>(x, xs_buf, s, hs);
      xin = xs_buf;
    }

    // qkv = xs @ qkv_w[i] + qkv_b[i]
    wmma_gemm_kernel<<<dim3(C3 / 64, Msub / 64), blk, 0, stream>>>(
        xin, qkv_w + (size_t)i * CDIM * C3, qkv_b + (size_t)i * C3, qkv_buf,
        Msub, C3, CDIM, nullptr, 0, 0);

    // attention (scores + softmax + head-mean + P@V)
    size_t smem = (size_t)(16 * QKP + 4 * 16 * QKP + 16 * 64 + 16 * (Ns + 4) +
                           16 * Ns + 128 + 32) * sizeof(float);
    float* attn_o = (i == 0) ? attn0 : (i == 1) ? attn1 : attn2;
    attn_kernel<<<dim3(Ns / 16, BATCH), blk, smem, stream>>>(qkv_buf, o_buf,
                                                             attn_o, Ns);

    // att_out = o @ proj_w[i] + proj_b[i]
    wmma_gemm_kernel<<<dim3(CDIM / 64, Msub / 64), blk, 0, stream>>>(
        o_buf, proj_w + (size_t)i * CDIM * CDIM, proj_b + (size_t)i * CDIM,
        att_buf, Msub, CDIM, CDIM, nullptr, 0, 0);

    // feat = att_out @ sp_w[i] + sp_b[i]; scale 0 writes w0*feat into fused,
    // scales 1/2 write feat then bilinear-upsample-accumulate w_i*up(feat).
    if (i == 0) {
      wmma_gemm_kernel<<<dim3(CDIM / 64, Msub / 64), blk, 0, stream>>>(
          att_buf, sp_w, sp_b, fused, Msub, CDIM, CDIM, fw, 0, 0);
    } else {
      wmma_gemm_kernel<<<dim3(CDIM / 64, Msub / 64), blk, 0, stream>>>(
          att_buf, sp_w + (size_t)i * CDIM * CDIM, sp_b + (size_t)i * CDIM,
          sm_buf, Msub, CDIM, CDIM, nullptr, 0, 0);
      int tot = BATCH * SEQ * CDIM;
      upsample_add_kernel<<<(tot + 255) / 256, 256, 0, stream>>>(
          sm_buf, fused, fw, i, hs);
    }
  }

  // out = fused @ out_w + out_b
  wmma_gemm_kernel<<<dim3(CDIM / 64, (BATCH * SEQ) / 64), blk, 0, stream>>>(
      fused, out_w, out_b, out, BATCH * SEQ, CDIM, CDIM, nullptr, 0, 0);
}